// prob_mlp_59777354826020
// MI455X (gfx1250) — compile-verified
//
#include <hip/hip_runtime.h>
#include <hip/hip_bf16.h>

// ---------------------------------------------------------------------------
// MI455X (gfx1250) implementation.
//   B=256, D_IN=784, WIDTH=256, DEPTH=10, NCLS=10
// Math note: y >= 0 invariantly (relu chain + nonneg residuals), so
// mask == 1 and the per-sample Jacobian J_b = W_d + I is batch-independent.
// => 10 SVDs total (one per depth), each done by one workgroup with the full
// 256x256 Gram matrix resident in CDNA5's 320KB LDS.
// All GEMMs use V_WMMA_F32_16X16X4_F32 (exact f32 — matches reference dtype).
// Activation stripes are staged into LDS via the Tensor Data Mover
// (TENSOR_LOAD_TO_LDS + S_WAIT_TENSORCNT), cutting activation re-reads 16x.
// ---------------------------------------------------------------------------

typedef __attribute__((ext_vector_type(2))) float        v2f;
typedef __attribute__((ext_vector_type(8))) float        v8f;
typedef __attribute__((ext_vector_type(4))) unsigned int v4u;
typedef __attribute__((ext_vector_type(8))) int          v8i;
typedef __attribute__((ext_vector_type(4))) int          v4i;

#define WIDTH   256
#define D_IN    784
#define DEPTH   10
#define NCLS    10
#define GPITCH  257          // LDS row pitch (floats) to avoid bank conflicts
#define NSWEEP  8            // cyclic Jacobi sweeps

__device__ __forceinline__ v8f wmma_f32(v2f a, v2f b, v8f c) {
    // 8 args: (neg_a, A, neg_b, B, c_mod, C, reuse_a, reuse_b)
    return __builtin_amdgcn_wmma_f32_16x16x4_f32(
        false, a, false, b, (short)0, c, false, false);
}

// TDM: 1-D tile DMA of `tile_floats` f32 from global `src` into LDS `lds_off`.
// D# layout per CDNA5 ISA ch.8: group0 {count=1, lds_addr, global_addr,
// type=2}; group1 {data_size=4B, tensor_dim0, tile_dim0, 1-D tile}.
// This toolchain exposes the 6-arg builtin: the extra int32x8 group is
// zero-filled (unused for a 1-D, non-cluster, non-gather descriptor).
__device__ __forceinline__ void tdm_load_1d(unsigned lds_off,
                                            const float* src,
                                            unsigned tensor_floats,
                                            unsigned tile_floats) {
    unsigned long long ga = (unsigned long long)(size_t)src;
    v4u g0;
    g0[0] = 1u;                                   // count=1, user mode
    g0[1] = lds_off;                              // lds_addr (byte offset)
    g0[2] = (unsigned)ga;                         // global_addr[31:0]
    g0[3] = (unsigned)((ga >> 32) & 0x1FFFFFFu)   // global_addr[56:32]
          | 0x80000000u;                          // type=2 at bits[127:126]
    v8i g1;
    g1[0] = 0x20000;                              // data_size=2 (4B), mask=0
    g1[1] = (int)((tensor_floats & 0xFFFFu) << 16);   // tensor_dim0[15:0]
    g1[2] = (int)(tensor_floats >> 16);               // tensor_dim0[31:16]
    g1[3] = (int)(tile_floats << 16);                 // tile_dim0
    g1[4] = 0;                                        // tile_dim1/2 = 0 (1-D)
    g1[5] = (int)tensor_floats;                       // dim0 stride (benign)
    g1[6] = 0;
    g1[7] = 0;
    v4i gz4 = {};                                 // groups 2/3: <=2-D tensor
    v8i gz8 = {};                                 // extra group (unused)
    __builtin_amdgcn_tensor_load_to_lds(g0, g1, gz4, gz4, gz8, 0);
}

// ------------------------------ first fc + relu ----------------------------
// y[256,256] = relu(x[256,784] @ W1[256,784]^T + b1)
// One WG per 16-row stripe: TDM-stage x stripe to LDS, 8 waves x 2 col tiles.
__global__ __launch_bounds__(256)
void fc1_kernel(const float* __restrict__ x, const float* __restrict__ W1,
                const float* __restrict__ b1, float* __restrict__ y) {
    __shared__ float As[16 * D_IN];               // 50KB activation stripe
    const int tid  = threadIdx.x;
    const int wave = tid >> 5, lane = tid & 31;
    const int half = lane >> 4, idx = lane & 15;
    const int row0 = blockIdx.x * 16;

    if (wave == 0) {
        tdm_load_1d((unsigned)(size_t)&As[0], x + row0 * D_IN,
                    256u * D_IN, 16u * D_IN);
        __builtin_amdgcn_s_wait_tensorcnt(0);
    }
    __syncthreads();

    for (int t = wave; t < 16; t += 8) {
        const int col0 = t * 16;
        v8f acc = {};
        for (int k0 = 0; k0 < D_IN; k0 += 4) {
            const int ka = k0 + 2 * half;
            v2f a; a.x = As[idx * D_IN + ka]; a.y = As[idx * D_IN + ka + 1];
            const float* bp = W1 + (col0 + idx) * D_IN + ka;   // B = W1^T
            v2f b; b.x = bp[0]; b.y = bp[1];
            acc = wmma_f32(a, b, acc);
        }
        const int col = col0 + idx;
        const float bias = b1[col];
        for (int v = 0; v < 8; ++v) {
            const int row = row0 + v + 8 * half;
            y[row * WIDTH + col] = fmaxf(acc[v] + bias, 0.0f);
        }
    }
}

// ------------------------------ residual block -----------------------------
// yout = relu(yin @ W^T + b) + yin ; A stripe + residual served from LDS.
__global__ __launch_bounds__(256)
void resblock_kernel(const float* __restrict__ yin, const float* __restrict__ W,
                     const float* __restrict__ bias, float* __restrict__ yout) {
    __shared__ float As[16 * WIDTH];              // 16KB activation stripe
    const int tid  = threadIdx.x;
    const int wave = tid >> 5, lane = tid & 31;
    const int half = lane >> 4, idx = lane & 15;
    const int row0 = blockIdx.x * 16;

    if (wave == 0) {
        tdm_load_1d((unsigned)(size_t)&As[0], yin + row0 * WIDTH,
                    256u * WIDTH, 16u * WIDTH);
        __builtin_amdgcn_s_wait_tensorcnt(0);
    }
    __syncthreads();

    for (int t = wave; t < 16; t += 8) {
        const int col0 = t * 16;
        v8f acc = {};
        for (int k0 = 0; k0 < WIDTH; k0 += 4) {
            const int ka = k0 + 2 * half;
            v2f a; a.x = As[idx * WIDTH + ka]; a.y = As[idx * WIDTH + ka + 1];
            const float* bp = W + (col0 + idx) * WIDTH + ka;
            v2f b; b.x = bp[0]; b.y = bp[1];
            acc = wmma_f32(a, b, acc);
        }
        const int col = col0 + idx;
        const float bs = bias[col];
        for (int v = 0; v < 8; ++v) {
            const int r = v + 8 * half;           // row within stripe
            yout[(row0 + r) * WIDTH + col] =
                fmaxf(acc[v] + bs, 0.0f) + As[r * WIDTH + col];
        }
    }
}

// ------------------------------ last fc ------------------------------------
// out[256,10] = y @ Wl[10,256]^T + bl   (one 16-wide tile, cols 10..15 masked)
__global__ __launch_bounds__(32)
void lastfc_kernel(const float* __restrict__ y, const float* __restrict__ Wl,
                   const float* __restrict__ bl, float* __restrict__ out) {
    const int lane = threadIdx.x;
    const int half = lane >> 4, idx = lane & 15;
    const int row0 = blockIdx.x * 16;
    v8f acc = {};
    for (int k0 = 0; k0 < WIDTH; k0 += 4) {
        const int ka = k0 + 2 * half;
        const float* ap = y + (row0 + idx) * WIDTH + ka;
        v2f a; a.x = ap[0]; a.y = ap[1];
        v2f b;
        if (idx < NCLS) {
            const float* bp = Wl + idx * WIDTH + ka;
            b.x = bp[0]; b.y = bp[1];
        } else { b.x = 0.0f; b.y = 0.0f; }
        acc = wmma_f32(a, b, acc);
    }
    if (idx < NCLS) {
        const float bs = bl[idx];
        for (int v = 0; v < 8; ++v) {
            const int row = row0 + v + 8 * half;
            out[row * NCLS + idx] = acc[v] + bs;
        }
    }
}

// ------------------------------ sigma kernel -------------------------------
// One workgroup per depth d. J = W_d + I.
//  Phase 1: G = J^T J into LDS via WMMA (8 waves x 32 tiles x 64 k-steps).
//  Phase 2: parallel cyclic Jacobi (round-robin pairing, 128 rotations/step).
//  Phase 3: sigma = sqrt(diag), bitonic sort descending, write [d][256].
__global__ __launch_bounds__(256)
void sigma_kernel(const float* __restrict__ Wb, float* __restrict__ sig_out) {
    extern __shared__ float G[];                 // [256][GPITCH] floats
    __shared__ float c_arr[128], s_arr[128];
    __shared__ int   p_arr[128], q_arr[128];
    __shared__ float sv[WIDTH];

    const int tid  = threadIdx.x;
    const int wave = tid >> 5, lane = tid & 31;
    const int half = lane >> 4, idx = lane & 15;
    const float* W = Wb + (size_t)blockIdx.x * WIDTH * WIDTH;

    // ---- Phase 1: Gram matrix via WMMA ----
    for (int tile = wave; tile < 256; tile += 8) {
        const int i0 = (tile >> 4) * 16;         // G row block
        const int j0 = (tile & 15) * 16;         // G col block
        v8f acc = {};
        for (int k0 = 0; k0 < WIDTH; k0 += 4) {
            const int ka = k0 + 2 * half;
            const int ca = i0 + idx, cb = j0 + idx;
            v2f a, b;
            // A = J^T tile: A[m][k] = J[k][i0+m] = W[k*256 + c] + (k==c)
            a.x = W[ ka      * WIDTH + ca] + ((ka     == ca) ? 1.0f : 0.0f);
            a.y = W[(ka + 1) * WIDTH + ca] + ((ka + 1 == ca) ? 1.0f : 0.0f);
            b.x = W[ ka      * WIDTH + cb] + ((ka     == cb) ? 1.0f : 0.0f);
            b.y = W[(ka + 1) * WIDTH + cb] + ((ka + 1 == cb) ? 1.0f : 0.0f);
            acc = wmma_f32(a, b, acc);
        }
        for (int v = 0; v < 8; ++v) {
            const int row = i0 + v + 8 * half;
            G[row * GPITCH + j0 + idx] = acc[v];
        }
    }
    __syncthreads();

    // ---- Phase 2: Jacobi sweeps ----
    for (int sweep = 0; sweep < NSWEEP; ++sweep) {
        for (int r = 0; r < 255; ++r) {
            if (tid < 128) {
                int p, q;
                if (tid == 0) { p = 255; q = r; }
                else {
                    p = (r + tid) % 255;
                    q = (r - tid + 510) % 255;
                }
                if (p > q) { int t = p; p = q; q = t; }
                const float app = G[p * GPITCH + p];
                const float aqq = G[q * GPITCH + q];
                const float apq = G[p * GPITCH + q];
                float c = 1.0f, s = 0.0f;
                if (apq != 0.0f) {
                    const float tau = (aqq - app) / (2.0f * apq);
                    const float t = (tau >= 0.0f ? 1.0f : -1.0f) /
                                    (fabsf(tau) + sqrtf(1.0f + tau * tau));
                    c = 1.0f / sqrtf(1.0f + t * t);
                    s = t * c;
                }
                c_arr[tid] = c; s_arr[tid] = s; p_arr[tid] = p; q_arr[tid] = q;
            }
            __syncthreads();
            // G <- G * R   (column rotations; disjoint column pairs)
            for (int k = 0; k < 128; ++k) {
                const float c = c_arr[k], s = s_arr[k];
                const int p = p_arr[k], q = q_arr[k];
                const float gp = G[tid * GPITCH + p];
                const float gq = G[tid * GPITCH + q];
                G[tid * GPITCH + p] = c * gp - s * gq;
                G[tid * GPITCH + q] = s * gp + c * gq;
            }
            __syncthreads();
            // G <- R^T * G (row rotations)
            for (int k = 0; k < 128; ++k) {
                const float c = c_arr[k], s = s_arr[k];
                const int p = p_arr[k], q = q_arr[k];
                const float gp = G[p * GPITCH + tid];
                const float gq = G[q * GPITCH + tid];
                G[p * GPITCH + tid] = c * gp - s * gq;
                G[q * GPITCH + tid] = s * gp + c * gq;
            }
            __syncthreads();
        }
    }

    // ---- Phase 3: singular values = sqrt(eigs), sort descending ----
    sv[tid] = sqrtf(fmaxf(G[tid * GPITCH + tid], 0.0f));
    __syncthreads();
    for (int size = 2; size <= WIDTH; size <<= 1) {
        for (int stride = size >> 1; stride > 0; stride >>= 1) {
            const int partner = tid ^ stride;
            if (partner > tid) {
                const bool desc = ((tid & size) == 0);
                const float a0 = sv[tid], b0 = sv[partner];
                const bool doswap = desc ? (a0 < b0) : (a0 > b0);
                if (doswap) { sv[tid] = b0; sv[partner] = a0; }
            }
            __syncthreads();
        }
    }
    sig_out[blockIdx.x * WIDTH + tid] = sv[tid];
}

// ------------------------------ sigma mean ---------------------------------
// all_sigma[b][i] = mean_d sigma[d][i]   (identical across b; broadcast)
__global__ __launch_bounds__(256)
void sigma_mean_kernel(const float* __restrict__ sig,
                       float* __restrict__ all_sigma) {
    const int i = threadIdx.x, b = blockIdx.x;
    float acc = 0.0f;
    for (int d = 0; d < DEPTH; ++d) acc += sig[d * WIDTH + i];
    all_sigma[b * WIDTH + i] = acc * (1.0f / DEPTH);
}

// ---------------------------------------------------------------------------
extern "C" void kernel_launch(void* const* d_in, const int* in_sizes, int n_in,
                              void* d_out, int out_size, void* d_ws, size_t ws_size,
                              hipStream_t stream) {
    (void)in_sizes; (void)n_in; (void)out_size; (void)ws_size;
    const float* x  = (const float*)d_in[0];
    const float* W1 = (const float*)d_in[1];
    const float* b1 = (const float*)d_in[2];
    const float* Wb = (const float*)d_in[3];
    const float* bb = (const float*)d_in[4];
    const float* Wl = (const float*)d_in[5];
    const float* bl = (const float*)d_in[6];

    float* out       = (float*)d_out;              // [256,10]
    float* all_sigma = out + 256 * NCLS;           // [256,256]

    float* ws  = (float*)d_ws;
    float* yA  = ws;                               // [256,256]
    float* yB  = ws + 256 * WIDTH;                 // [256,256]
    float* sig = ws + 2 * 256 * WIDTH;             // [DEPTH,256]

    // Spectral path (independent of forward pass): 10 WGs, 263KB LDS each.
    const size_t lds_bytes = (size_t)WIDTH * GPITCH * sizeof(float);
    sigma_kernel<<<DEPTH, 256, lds_bytes, stream>>>(Wb, sig);

    // Forward chain (TDM-staged activation stripes).
    fc1_kernel<<<16, 256, 0, stream>>>(x, W1, b1, yA);
    float* yi = yA; float* yo = yB;
    for (int d = 0; d < DEPTH; ++d) {
        resblock_kernel<<<16, 256, 0, stream>>>(
            yi, Wb + (size_t)d * WIDTH * WIDTH, bb + d * WIDTH, yo);
        float* t = yi; yi = yo; yo = t;
    }
    lastfc_kernel<<<16, 32, 0, stream>>>(yi, Wl, bl, out);

    sigma_mean_kernel<<<256, 256, 0, stream>>>(sig, all_sigma);
}